// RoIAlign_3685081940467
// MI455X (gfx1250) — compile-verified
//
#include <hip/hip_runtime.h>

// RoI-Align (avg, aligned, sampling_ratio=2) as separable y-interp + WMMA GEMM.
// x: (2,256,200,200) f32, rois: (1000,5) f32 -> out: (1000,256,7,7) f32.
//
// Per (roi, ph) block:
//   U[x,c]   = sum over the bin's 2 y-samples x 2 corners of wy * X[y, xs+x, c]
//              (coalesced row-segment loads, y-interp fused, staged to LDS)
//   out[pw,c]= sum_x A[pw,x] * U[x,c]   via V_WMMA_F32_16X16X4_F32, K = 4*kcnt
// A folds the 1/4 sample-average and the x bilinear weights (dense 7 x span).

#define OUT_HW  7
#define CCH     256
#define HH      200
#define WW      200
#define XMAX    64      // max x-span of a RoI's sample footprint (<= 52 by construction)
#define CPAD    129     // LDS row pitch for U (odd -> conflict-free staging)
#define CCHUNK  128     // channels staged per pass (8 waves x 16-ch N-tiles)

typedef float v2f __attribute__((ext_vector_type(2)));
typedef float v8f __attribute__((ext_vector_type(8)));

__global__ __launch_bounds__(256)
void roi_align_wmma(const float* __restrict__ in, const float* __restrict__ rois,
                    float* __restrict__ out)
{
    const int ph   = blockIdx.x;   // 0..6
    const int r    = blockIdx.y;   // roi
    const int tid  = threadIdx.x;
    const int lane = tid & 31;
    const int wave = tid >> 5;

    __shared__ float A_s[16 * XMAX];        // 7x64 pooling/interp weights, zero-padded to 16 rows
    __shared__ float U_s[XMAX * CPAD];      // y-interpolated rows: U[x][c]
    __shared__ int   s_xlo[14], s_xhi[14];
    __shared__ float s_wxlo[14], s_wxhi[14];
    __shared__ int   s_yi[4];
    __shared__ float s_wy[4];
    __shared__ int   s_b, s_xs, s_kcnt;

    // ---- per-RoI metadata (wave-uniform, computed once per block) ----
    for (int i = tid; i < 16 * XMAX; i += 256) A_s[i] = 0.0f;

    if (tid < 14) {                               // 14 x-sample columns
        const float x1 = rois[r * 5 + 1] * 0.25f - 0.5f;
        const float x2 = rois[r * 5 + 3] * 0.25f - 0.5f;
        const float bw = (x2 - x1) * (1.0f / OUT_HW);
        const float sx = x1 + ((float)tid + 0.5f) * (bw * 0.5f);
        const bool  valid = (sx >= -1.0f) && (sx <= (float)WW);
        const float sc = fmaxf(sx, 0.0f);
        int lo = (int)floorf(sc); if (lo > WW - 1) lo = WW - 1;
        int hi = lo + 1;          if (hi > WW - 1) hi = WW - 1;
        const float fr = (sc >= (float)(WW - 1)) ? 0.0f : (sc - (float)lo);
        s_xlo[tid] = lo;  s_xhi[tid] = hi;
        s_wxlo[tid] = valid ? (1.0f - fr) : 0.0f;
        s_wxhi[tid] = valid ? fr          : 0.0f;
    } else if (tid < 16) {                        // the 2 y-samples of this ph bin
        const int   s  = tid - 14;
        const float y1 = rois[r * 5 + 2] * 0.25f - 0.5f;
        const float y2 = rois[r * 5 + 4] * 0.25f - 0.5f;
        const float bh = (y2 - y1) * (1.0f / OUT_HW);
        const float sy = y1 + ((float)(2 * ph + s) + 0.5f) * (bh * 0.5f);
        const bool  valid = (sy >= -1.0f) && (sy <= (float)HH);
        const float sc = fmaxf(sy, 0.0f);
        int lo = (int)floorf(sc); if (lo > HH - 1) lo = HH - 1;
        int hi = lo + 1;          if (hi > HH - 1) hi = HH - 1;
        const float fr = (sc >= (float)(HH - 1)) ? 0.0f : (sc - (float)lo);
        s_yi[2 * s] = lo;  s_yi[2 * s + 1] = hi;
        s_wy[2 * s]     = valid ? (1.0f - fr) : 0.0f;
        s_wy[2 * s + 1] = valid ? fr          : 0.0f;
    } else if (tid == 16) {
        s_b = (int)rois[r * 5 + 0];
    }
    __syncthreads();

    if (tid == 0) {                               // build dense A (7 x span)
        const int xs = s_xlo[0];                  // sample x is monotone -> min at j=0
        s_xs = xs;
        const int span = s_xhi[13] - xs + 1;      // <= 52
        s_kcnt = (span + 3) >> 2;                 // K-steps of 4
        #pragma unroll
        for (int j = 0; j < 14; ++j) {
            const int pw = j >> 1;
            A_s[pw * XMAX + (s_xlo[j] - xs)] += 0.25f * s_wxlo[j];
            A_s[pw * XMAX + (s_xhi[j] - xs)] += 0.25f * s_wxhi[j];
        }
    }
    __syncthreads();

    const int   b    = s_b;
    const int   xs   = s_xs;
    const int   kcnt = s_kcnt;
    const int   xlim = kcnt << 2;                 // only U[0 .. 4*kcnt) is ever read
    const int   y0 = s_yi[0], y1i = s_yi[1], y2i = s_yi[2], y3 = s_yi[3];
    const float w0 = s_wy[0], w1 = s_wy[1], w2 = s_wy[2], w3 = s_wy[3];

    const int x  = tid & 63;                      // staging: lanes sweep x (coalesced)
    const int cg = tid >> 6;                      // 4 channel groups
    int gx = xs + x; if (gx > WW - 1) gx = WW - 1;

    for (int chunk = 0; chunk < CCH; chunk += CCHUNK) {
        __syncthreads();                          // U_s safe to overwrite

        // ---- stage: coalesced row loads + y-interp into LDS (footprint only) ----
        if (x < xlim) {
            for (int ci = cg; ci < CCHUNK; ci += 4) {
                const int c = chunk + ci;
                const float* p = in + (size_t)(b * CCH + c) * (size_t)(HH * WW);
                const float v0 = p[y0  * WW + gx];
                const float v1 = p[y1i * WW + gx];
                const float v2 = p[y2i * WW + gx];
                const float v3 = p[y3  * WW + gx];
                if (ci + 4 < CCHUNK) {            // warm L0 for next channel iteration
                    const float* pn = p + 4 * (size_t)(HH * WW);
                    __builtin_prefetch(pn + y0  * WW + gx, 0, 1);
                    __builtin_prefetch(pn + y2i * WW + gx, 0, 1);
                }
                U_s[x * CPAD + ci] = w0 * v0 + w1 * v1 + w2 * v2 + w3 * v3;
            }
        }
        __syncthreads();

        // ---- WMMA: out(16x16) = A(16xK) * U(Kx16), one 16-ch N-tile per wave ----
        const int nt      = wave;                 // 8 waves x 16 ch = 128-ch chunk
        const int m       = lane & 15;
        const int kparity = (lane >> 4) << 1;     // lanes 0-15 -> K0/K1, 16-31 -> K2/K3
        v8f acc = {};
        for (int kk = 0; kk < kcnt; ++kk) {
            const int kb = 4 * kk + kparity;
            v2f a, bm;
            a.x  = A_s[m * XMAX + kb];
            a.y  = A_s[m * XMAX + kb + 1];
            bm.x = U_s[kb       * CPAD + nt * 16 + m];
            bm.y = U_s[(kb + 1) * CPAD + nt * 16 + m];
            acc = __builtin_amdgcn_wmma_f32_16x16x4_f32(
                false, a, false, bm, (short)0, acc, false, false);
        }

        // ---- D layout: lanes 0-15 hold M=0..7 in v0..v7; pw<7 valid ----
        if (lane < 16) {
            const int c = chunk + nt * 16 + lane;
            float* o = out + ((size_t)r * CCH + (size_t)c) * (size_t)(OUT_HW * OUT_HW)
                           + (size_t)ph * OUT_HW;
            #pragma unroll
            for (int pw = 0; pw < OUT_HW; ++pw) o[pw] = acc[pw];
        }
    }
}

extern "C" void kernel_launch(void* const* d_in, const int* in_sizes, int n_in,
                              void* d_out, int out_size, void* d_ws, size_t ws_size,
                              hipStream_t stream) {
    const float* in   = (const float*)d_in[0];
    const float* rois = (const float*)d_in[1];
    float*       out  = (float*)d_out;
    const int R = in_sizes[1] / 5;                // 1000
    dim3 grid(OUT_HW, R);
    roi_align_wmma<<<grid, 256, 0, stream>>>(in, rois, out);
}